// GinNet_64991445123442
// MI455X (gfx1250) — compile-verified
//
#include <hip/hip_runtime.h>

// ---------------------------------------------------------------------------
// GIN (3x GINConv + linear head) for MI455X / gfx1250, fp32 end-to-end.
// MLP GEMMs use V_WMMA_F32_16X16X4_F32 (wave32, 16x16 f32 tiles, K=4).
// Aggregation is scatter-add with global_atomic_add_f32 (L2-resident dsts).
// K=10 tail handled branchlessly: clamped A address + select-masked B rows
// (A * 0 == 0 for finite inputs), so no EXEC-toggling around WMMA feeds.
// ---------------------------------------------------------------------------

typedef __attribute__((ext_vector_type(2))) float v2f;
typedef __attribute__((ext_vector_type(8))) float v8f;

#define N_NODES_C 50000
#define N_EDGES_C 800000
#define F_IN_C    10
#define HID_C     64

// ---------------- scatter-add aggregation ----------------------------------

__global__ void gin_scatter_f10(const float* __restrict__ x,
                                const int* __restrict__ src,
                                const int* __restrict__ dst,
                                float* __restrict__ agg, int nE) {
  int e = blockIdx.x * blockDim.x + threadIdx.x;
  if (e >= nE) return;
  const int s = src[e], d = dst[e];
  const float* xs = x + (long)s * F_IN_C;
  float* ad = agg + (long)d * F_IN_C;
#pragma unroll
  for (int f = 0; f < F_IN_C; ++f) atomicAdd(ad + f, xs[f]);
}

__global__ void gin_scatter_f64(const float* __restrict__ h,
                                const int* __restrict__ src,
                                const int* __restrict__ dst,
                                float* __restrict__ agg, int nE) {
  int idx = blockIdx.x * blockDim.x + threadIdx.x;
  int e = idx >> 4;                 // 16 threads per edge, 4 feats each
  if (e >= nE) return;
  const int c = (idx & 15) * 4;
  const int s = src[e], d = dst[e];
  const float4 v = *(const float4*)(h + (long)s * HID_C + c);
  float* ad = agg + (long)d * HID_C + c;
  atomicAdd(ad + 0, v.x);
  atomicAdd(ad + 1, v.y);
  atomicAdd(ad + 2, v.z);
  atomicAdd(ad + 3, v.w);
}

// ---------------- fused GIN MLP: relu(relu(h@W1+b1)@W2+b2) -----------------
// h = (1+eps)*x + agg fused into the A-operand load of GEMM1.
// Block: 128 threads = 4 waves; each wave computes a 16x64 output slab.
// A/B VGPR layout for V_WMMA_F32_16X16X4_F32 (ISA 7.12.2):
//   lanes 0-15  : K = k0+0 (v0), k0+1 (v1),  M/N = lane
//   lanes 16-31 : K = k0+2 (v0), k0+3 (v1),  M/N = lane-16
// C/D: vgpr r -> row = r + 8*(lane>=16), col = (lane&15) + 16*tile

template <int KIN>
__global__ __launch_bounds__(128) void gin_mlp_wmma(
    const float* __restrict__ xin, const float* __restrict__ agg,
    const float* __restrict__ epsP,
    const float* __restrict__ W1, const float* __restrict__ b1,
    const float* __restrict__ W2, const float* __restrict__ b2,
    float* __restrict__ out, int nNodes) {
  __shared__ float sh[64][HID_C + 1];   // +1 pad: conflict-free col reads

  const int lane = threadIdx.x & 31;
  const int wave = threadIdx.x >> 5;
  const int r    = lane & 15;
  const int half = lane >> 4;           // 0 or 1
  const int rowBase = blockIdx.x * 64 + wave * 16;
  const float onePlusEps = 1.0f + epsP[0];

  v8f acc[4] = {{}, {}, {}, {}};

  // ---- GEMM1: [16 x KIN] @ [KIN x 64] ----
  const int  arow  = min(rowBase + r, nNodes - 1);  // clamp tail (stores guarded)
  const long abase = (long)arow * KIN;
  constexpr int KSTEPS1 = (KIN + 3) / 4;
#pragma unroll
  for (int ks = 0; ks < KSTEPS1; ++ks) {
    const int kk = ks * 4 + 2 * half;
    // A: clamped unconditional loads; out-of-range K columns are garbage but
    // get multiplied by B == 0 below, contributing exactly 0.
    const int ka0 = (kk + 0 < KIN) ? kk + 0 : KIN - 1;
    const int ka1 = (kk + 1 < KIN) ? kk + 1 : KIN - 1;
    v2f a;
    a.x = onePlusEps * xin[abase + ka0] + agg[abase + ka0];
    a.y = onePlusEps * xin[abase + ka1] + agg[abase + ka1];
#pragma unroll
    for (int t = 0; t < 4; ++t) {
      const int n = t * 16 + r;
      // B: clamped unconditional load, then branchless zero-mask (v_cndmask).
      float b0 = W1[(long)ka0 * HID_C + n];
      float b1v = W1[(long)ka1 * HID_C + n];
      v2f b;
      b.x = (kk + 0 < KIN) ? b0 : 0.0f;
      b.y = (kk + 1 < KIN) ? b1v : 0.0f;
      acc[t] = __builtin_amdgcn_wmma_f32_16x16x4_f32(
          false, a, false, b, (short)0, acc[t], false, false);
    }
  }

  // bias + ReLU -> LDS (each wave writes only its own 16-row slab)
#pragma unroll
  for (int t = 0; t < 4; ++t) {
    const int n = t * 16 + r;
    const float bv = b1[n];
#pragma unroll
    for (int rr = 0; rr < 8; ++rr) {
      sh[wave * 16 + rr + 8 * half][n] = fmaxf(acc[t][rr] + bv, 0.0f);
    }
    acc[t] = (v8f){};
  }
  __syncthreads();

  // ---- GEMM2: [16 x 64] @ [64 x 64] ----
#pragma unroll
  for (int ks = 0; ks < HID_C / 4; ++ks) {
    const int kk = ks * 4 + 2 * half;
    v2f a;
    a.x = sh[wave * 16 + r][kk + 0];
    a.y = sh[wave * 16 + r][kk + 1];
#pragma unroll
    for (int t = 0; t < 4; ++t) {
      const int n = t * 16 + r;
      v2f b;
      b.x = W2[(long)(kk + 0) * HID_C + n];
      b.y = W2[(long)(kk + 1) * HID_C + n];
      acc[t] = __builtin_amdgcn_wmma_f32_16x16x4_f32(
          false, a, false, b, (short)0, acc[t], false, false);
    }
  }

  // bias + outer ReLU -> global (guard tail rows; all WMMA already done)
#pragma unroll
  for (int t = 0; t < 4; ++t) {
    const int n = t * 16 + r;
    const float bv = b2[n];
#pragma unroll
    for (int rr = 0; rr < 8; ++rr) {
      const int row = rowBase + rr + 8 * half;
      if (row < nNodes) out[(long)row * HID_C + n] = fmaxf(acc[t][rr] + bv, 0.0f);
    }
  }
}

// ---------------- final 64 -> 1 linear -------------------------------------

__global__ void gin_final_linear(const float* __restrict__ h,
                                 const float* __restrict__ Wf,
                                 const float* __restrict__ bf,
                                 float* __restrict__ out, int n) {
  int i = blockIdx.x * blockDim.x + threadIdx.x;
  if (i >= n) return;
  const float* hr = h + (long)i * HID_C;
  float s = bf[0];
#pragma unroll
  for (int f = 0; f < HID_C; ++f) s += hr[f] * Wf[f];
  out[i] = s;
}

// ---------------- driver ----------------------------------------------------

extern "C" void kernel_launch(void* const* d_in, const int* in_sizes, int n_in,
                              void* d_out, int out_size, void* d_ws, size_t ws_size,
                              hipStream_t stream) {
  (void)in_sizes; (void)n_in; (void)out_size; (void)ws_size;

  const float* x    = (const float*)d_in[0];
  const int*   ei   = (const int*)d_in[1];
  const int*   src  = ei;
  const int*   dst  = ei + N_EDGES_C;
  const float* eps1 = (const float*)d_in[2];
  const float* W1a  = (const float*)d_in[3];
  const float* b1a  = (const float*)d_in[4];
  const float* W1b  = (const float*)d_in[5];
  const float* b1b  = (const float*)d_in[6];
  const float* eps2 = (const float*)d_in[7];
  const float* W2a  = (const float*)d_in[8];
  const float* b2a  = (const float*)d_in[9];
  const float* W2b  = (const float*)d_in[10];
  const float* b2b  = (const float*)d_in[11];
  const float* eps3 = (const float*)d_in[12];
  const float* W3a  = (const float*)d_in[13];
  const float* b3a  = (const float*)d_in[14];
  const float* W3b  = (const float*)d_in[15];
  const float* b3b  = (const float*)d_in[16];
  const float* Wf   = (const float*)d_in[17];
  const float* bf   = (const float*)d_in[18];

  float* agg = (float*)d_ws;
  float* hA  = agg + (size_t)N_NODES_C * HID_C;
  float* hB  = hA  + (size_t)N_NODES_C * HID_C;

  const int mlpBlocks  = (N_NODES_C + 63) / 64;
  const int sc10Blocks = (N_EDGES_C + 255) / 256;
  const int sc64Blocks = (N_EDGES_C * 16 + 255) / 256;

  // conv1 : F_IN -> 64
  hipMemsetAsync(agg, 0, (size_t)N_NODES_C * F_IN_C * sizeof(float), stream);
  gin_scatter_f10<<<sc10Blocks, 256, 0, stream>>>(x, src, dst, agg, N_EDGES_C);
  gin_mlp_wmma<F_IN_C><<<mlpBlocks, 128, 0, stream>>>(
      x, agg, eps1, W1a, b1a, W1b, b1b, hA, N_NODES_C);

  // conv2 : 64 -> 64
  hipMemsetAsync(agg, 0, (size_t)N_NODES_C * HID_C * sizeof(float), stream);
  gin_scatter_f64<<<sc64Blocks, 256, 0, stream>>>(hA, src, dst, agg, N_EDGES_C);
  gin_mlp_wmma<HID_C><<<mlpBlocks, 128, 0, stream>>>(
      hA, agg, eps2, W2a, b2a, W2b, b2b, hB, N_NODES_C);

  // conv3 : 64 -> 64
  hipMemsetAsync(agg, 0, (size_t)N_NODES_C * HID_C * sizeof(float), stream);
  gin_scatter_f64<<<sc64Blocks, 256, 0, stream>>>(hB, src, dst, agg, N_EDGES_C);
  gin_mlp_wmma<HID_C><<<mlpBlocks, 128, 0, stream>>>(
      hB, agg, eps3, W3a, b3a, W3b, b3b, hA, N_NODES_C);

  // head : 64 -> 1
  gin_final_linear<<<(N_NODES_C + 255) / 256, 256, 0, stream>>>(
      hA, Wf, bf, (float*)d_out, N_NODES_C);
}